// TrustworthyNet_classfier_51402168598565
// MI455X (gfx1250) — compile-verified
//
#include <hip/hip_runtime.h>

typedef __attribute__((ext_vector_type(2))) float v2f;
typedef __attribute__((ext_vector_type(4))) float v4f;
typedef __attribute__((ext_vector_type(8))) float v8f;

#define N_SAMP 8192
#define FEAT   2048
#define KCLS   256
#define NVIEW  3
#define GAMMA  0.5f

// ---------------------------------------------------------------------------
// GEMM: C[M x Ncls] = beta*C + scale * A[M x Kd] * opB
//  TRANS_B = true : B is [Ncls x Kd] row-major, use B^T  (C = A @ B^T)
//  TRANS_B = false: B is [Kd x Ncls] row-major            (C = A @ B)
// Block tile 128(M) x 64(N); 256 threads = 8 waves, each wave owns 32x32
// (2x2 WMMA 16x16 tiles). K staged through LDS in 32-wide slabs with
// double-buffered LDS + register prefetch so global loads for slab k+1
// overlap the 32 WMMAs of slab k. One barrier per slab.
//
// LDS B layout interleaves K-pairs: element (k,n) at (k>>1)*128 + 2n + (k&1)
// so a lane's (kb, kb+1) B fragment is one ds_load_b64 (matches the WMMA
// 4x16 B layout: lanes 0-15/16-31 hold K{0,1}/K{2,3} for N=0..15).
//
// Requires: M%128==0, Ncls%64==0, Kd%32==0 (true for all shapes here).
// ---------------------------------------------------------------------------
template <bool TRANS_B>
__global__ __launch_bounds__(256)
void gemm_f32_wmma(const float* __restrict__ A, const float* __restrict__ B,
                   float* __restrict__ C, int M, int Ncls, int Kd,
                   int ldA, int ldB, int ldC, float beta, float scale)
{
    __shared__ float As[2][128 * 32];   // [buf][row][k]
    __shared__ float Bs[2][16 * 128];   // [buf][(k>>1)][n*2 + (k&1)]

    const int tid  = threadIdx.x;
    const int lane = tid & 31;
    const int wave = tid >> 5;
    const int wm0  = (wave >> 1) * 32;   // wave tile origin (M) in block tile
    const int wn0  = (wave & 1) * 32;    // wave tile origin (N) in block tile
    const int half = lane >> 4;          // 0: lanes 0-15, 1: lanes 16-31
    const int l16  = lane & 15;
    const int m0   = blockIdx.y * 128;
    const int n0   = blockIdx.x * 64;

    // staging coordinates
    const int ar_ = tid >> 3;            // A/B(NT) row base (0..31), +32*i
    const int ac4 = (tid & 7) * 4;       // k-offset within slab (A & B-NT)
    const int bk_ = tid >> 4;            // B(NN) k base (0..15), +16*i
    const int bc4 = (tid & 15) * 4;      // B(NN) n-offset

    v8f acc[2][2];
    #pragma unroll
    for (int mi = 0; mi < 2; ++mi)
        #pragma unroll
        for (int ni = 0; ni < 2; ++ni)
            acc[mi][ni] = v8f{};

    v4f pa[4], pb[2];
    auto prefetch = [&](int kk) {
        #pragma unroll
        for (int i = 0; i < 4; ++i)
            pa[i] = *(const v4f*)&A[(long)(m0 + ar_ + i * 32) * ldA + kk + ac4];
        if (TRANS_B) {
            #pragma unroll
            for (int i = 0; i < 2; ++i)
                pb[i] = *(const v4f*)&B[(long)(n0 + ar_ + i * 32) * ldB + kk + ac4];
        } else {
            #pragma unroll
            for (int i = 0; i < 2; ++i)
                pb[i] = *(const v4f*)&B[(long)(kk + bk_ + i * 16) * ldB + n0 + bc4];
        }
    };

    prefetch(0);
    int buf = 0;
    for (int kk = 0; kk < Kd; kk += 32) {
        // ---- commit prefetched registers to LDS buffer `buf`
        #pragma unroll
        for (int i = 0; i < 4; ++i)
            *(v4f*)&As[buf][(ar_ + i * 32) * 32 + ac4] = pa[i];
        if (TRANS_B) {
            #pragma unroll
            for (int i = 0; i < 2; ++i) {
                int n = ar_ + i * 32;
                v4f v = pb[i];
                *(v2f*)&Bs[buf][(ac4 >> 1) * 128 + n * 2]       = v2f{v.x, v.y};
                *(v2f*)&Bs[buf][((ac4 >> 1) + 1) * 128 + n * 2] = v2f{v.z, v.w};
            }
        } else {
            #pragma unroll
            for (int i = 0; i < 2; ++i) {
                int k = bk_ + i * 16;
                v4f v = pb[i];
                float* bp = &Bs[buf][(k >> 1) * 128 + (k & 1)];
                bp[(bc4 + 0) * 2] = v.x;
                bp[(bc4 + 1) * 2] = v.y;
                bp[(bc4 + 2) * 2] = v.z;
                bp[(bc4 + 3) * 2] = v.w;
            }
        }
        __syncthreads();

        // ---- issue next slab's global loads; they complete while we WMMA
        if (kk + 32 < Kd) prefetch(kk + 32);

        // ---- 8 k-steps of 4 over the 32-wide slab
        #pragma unroll
        for (int kq = 0; kq < 8; ++kq) {
            const int kb  = kq * 4 + half * 2;  // lane-half picks K{0,1}/K{2,3}
            const int kb2 = kq * 2 + half;
            v2f afrag[2], bfrag[2];
            #pragma unroll
            for (int mi = 0; mi < 2; ++mi)
                afrag[mi] = *(const v2f*)&As[buf][(wm0 + mi * 16 + l16) * 32 + kb];
            #pragma unroll
            for (int ni = 0; ni < 2; ++ni)
                bfrag[ni] = *(const v2f*)&Bs[buf][kb2 * 128 + (wn0 + ni * 16 + l16) * 2];
            #pragma unroll
            for (int mi = 0; mi < 2; ++mi)
                #pragma unroll
                for (int ni = 0; ni < 2; ++ni)
                    acc[mi][ni] = __builtin_amdgcn_wmma_f32_16x16x4_f32(
                        false, afrag[mi], false, bfrag[ni],
                        (short)0, acc[mi][ni], false, false);
        }
        buf ^= 1;
    }

    // ---- epilogue: C/D layout: VGPR v -> M = v + 8*half, N = l16
    #pragma unroll
    for (int mi = 0; mi < 2; ++mi)
        #pragma unroll
        for (int ni = 0; ni < 2; ++ni)
            #pragma unroll
            for (int v = 0; v < 8; ++v) {
                int gr = m0 + wm0 + mi * 16 + v + half * 8;
                int gc = n0 + wn0 + ni * 16 + l16;
                long idx = (long)gr * ldC + gc;
                C[idx] = beta * C[idx] + scale * acc[mi][ni][v];
            }
}

// ---------------------------------------------------------------------------
// Activation: selu(u - th) - selu(-u - th)
// ---------------------------------------------------------------------------
__device__ __forceinline__ float selu_f(float x) {
    const float SC = 1.0507009873554805f;
    const float AL = 1.6732632423543772f;
    return x > 0.f ? SC * x : SC * AL * (__expf(x) - 1.f);
}
__device__ __forceinline__ float act_f(float u, float th) {
    return selu_f(u - th) - selu_f(-u - th);
}

__global__ __launch_bounds__(256)
void epilogue_z(const float* __restrict__ acc, const float* __restrict__ bz,
                const float* __restrict__ theta, float* __restrict__ zinit,
                float* __restrict__ outz, float* __restrict__ outa)
{
    long i = (long)blockIdx.x * 256 + threadIdx.x;
    float th = theta[0];
    float z = act_f(acc[i] + bz[i & (KCLS - 1)], th);
    zinit[i] = z;
    outz[i]  = z;
    outa[i]  = z + 1.f;
}

__global__ __launch_bounds__(256)
void epilogue_h(const float* __restrict__ acc, const float* __restrict__ Sb,
                const float* __restrict__ Ub, const float* __restrict__ theta,
                float* __restrict__ outz1, float* __restrict__ outz2,
                float* __restrict__ outa1, float* __restrict__ outa2)
{
    long i = (long)blockIdx.x * 256 + threadIdx.x;
    float th = theta[0];
    int c = (int)(i & (KCLS - 1));
    float a = act_f(acc[i] + Sb[c] + Ub[c], th);
    outz1[i] = a;  outz2[i] = a;          // identical block-0 / block-1 slots
    outa1[i] = a + 1.f;  outa2[i] = a + 1.f;
}

// ---------------------------------------------------------------------------
// DS_Combin fold over the 9 alpha sets. One wave32 per row; each lane holds
// 8 of the 256 classes. Uses sum(b) = (S-K)/S and an analytic running-S
// update, so each fold needs only 2 cross-lane reductions.
// ---------------------------------------------------------------------------
__global__ __launch_bounds__(256)
void ds_combine(const float* __restrict__ outa,
                float* __restrict__ evi, float* __restrict__ alpha_out)
{
    const int wave = threadIdx.x >> 5;
    const int lane = threadIdx.x & 31;
    const long row = (long)blockIdx.x * 8 + wave;
    const float Kf = (float)KCLS;

    float ar[8];
    float s = 0.f;
    #pragma unroll
    for (int p = 0; p < 8; ++p) {
        ar[p] = outa[row * KCLS + lane + p * 32];
        s += ar[p];
    }
    #pragma unroll
    for (int off = 16; off > 0; off >>= 1) s += __shfl_xor(s, off, 32);
    float S1 = s;

    for (int slot = 1; slot < 9; ++slot) {
        const float* a2p = outa + (long)slot * N_SAMP * KCLS + row * KCLS;
        float a2[8];
        float s2 = 0.f;
        #pragma unroll
        for (int p = 0; p < 8; ++p) {
            a2[p] = a2p[lane + p * 32];
            s2 += a2[p];
        }
        #pragma unroll
        for (int off = 16; off > 0; off >>= 1) s2 += __shfl_xor(s2, off, 32);
        const float S2 = s2;

        const float u1 = Kf / S1, u2 = Kf / S2;
        float b1[8], b2[8], d = 0.f;
        #pragma unroll
        for (int p = 0; p < 8; ++p) {
            b1[p] = (ar[p] - 1.f) / S1;
            b2[p] = (a2[p] - 1.f) / S2;
            d += b1[p] * b2[p];
        }
        #pragma unroll
        for (int off = 16; off > 0; off >>= 1) d += __shfl_xor(d, off, 32);
        const float sumb1b2 = d;

        const float sumb1 = (S1 - Kf) / S1;
        const float sumb2 = (S2 - Kf) / S2;
        const float Cc    = sumb1 * sumb2 - sumb1b2;
        const float denom = 1.f - Cc;
        const float u_a   = u1 * u2 / denom;
        const float S_a   = Kf / u_a;
        #pragma unroll
        for (int p = 0; p < 8; ++p) {
            float b_a = (b1[p] * b2[p] + b1[p] * u2 + b2[p] * u1) / denom;
            ar[p] = b_a * S_a + 1.f;
        }
        const float sum_ba = (sumb1b2 + sumb1 * u2 + sumb2 * u1) / denom;
        S1 = sum_ba * S_a + Kf;   // analytic sum of updated alphas
    }

    #pragma unroll
    for (int p = 0; p < 8; ++p) {
        float v = ar[p];
        long idx = row * KCLS + lane + p * 32;
        alpha_out[idx] = v;
        evi[idx]       = v - 1.f;
    }
}

// ---------------------------------------------------------------------------
// Launch. Input order (setup_inputs dict order):
//   per view j: [8j+0]=f, [8j+1]=lap, [8j+2]=Wz, [8j+3]=bz,
//               [8j+4]=Sw, [8j+5]=Sb, [8j+6]=Uw, [8j+7]=Ub ; [24]=theta
// Output flat: out_z[9][N][K] | evi[N][K] | out_a[9][N][K] | alpha[N][K]
// ---------------------------------------------------------------------------
extern "C" void kernel_launch(void* const* d_in, const int* in_sizes, int n_in,
                              void* d_out, int out_size, void* d_ws, size_t ws_size,
                              hipStream_t stream)
{
    (void)in_sizes; (void)n_in; (void)out_size; (void)ws_size;
    const long NK = (long)N_SAMP * KCLS;

    float* acc   = (float*)d_ws;        // [N,K] pre-activation accumulator
    float* zinit = acc + NK;            // [N,K] current view's z_init

    float* out    = (float*)d_out;
    float* outz   = out;                // 9*NK
    float* evi    = out + 9 * NK;       // NK
    float* outa   = out + 10 * NK;      // 9*NK
    float* alpha  = out + 19 * NK;      // NK

    const float* theta = (const float*)d_in[24];

    dim3 blk(256);
    dim3 ggrid(KCLS / 64, N_SAMP / 128);
    int  egrid = (int)(NK / 256);

    for (int j = 0; j < NVIEW; ++j) {
        const float* f   = (const float*)d_in[j * 8 + 0];
        const float* lap = (const float*)d_in[j * 8 + 1];
        const float* Wz  = (const float*)d_in[j * 8 + 2];
        const float* bz  = (const float*)d_in[j * 8 + 3];
        const float* Sw  = (const float*)d_in[j * 8 + 4];
        const float* Sb  = (const float*)d_in[j * 8 + 5];
        const float* Uw  = (const float*)d_in[j * 8 + 6];
        const float* Ub  = (const float*)d_in[j * 8 + 7];

        // z_pre = f @ Wz^T
        gemm_f32_wmma<true><<<ggrid, blk, 0, stream>>>(
            f, Wz, acc, N_SAMP, KCLS, FEAT, FEAT, FEAT, KCLS, 0.f, 1.f);
        epilogue_z<<<egrid, blk, 0, stream>>>(
            acc, bz, theta, zinit, outz + j * NK, outa + j * NK);

        // h = z@Sw^T + f@Uw^T - gamma*lap@z   (identical for both blocks)
        gemm_f32_wmma<true><<<ggrid, blk, 0, stream>>>(
            zinit, Sw, acc, N_SAMP, KCLS, KCLS, KCLS, KCLS, KCLS, 0.f, 1.f);
        gemm_f32_wmma<true><<<ggrid, blk, 0, stream>>>(
            f, Uw, acc, N_SAMP, KCLS, FEAT, FEAT, FEAT, KCLS, 1.f, 1.f);
        gemm_f32_wmma<false><<<ggrid, blk, 0, stream>>>(
            lap, zinit, acc, N_SAMP, KCLS, N_SAMP, N_SAMP, KCLS, KCLS, 1.f, -GAMMA);
        epilogue_h<<<egrid, blk, 0, stream>>>(
            acc, Sb, Ub, theta,
            outz + (3 + j) * NK, outz + (6 + j) * NK,
            outa + (3 + j) * NK, outa + (6 + j) * NK);
    }

    ds_combine<<<N_SAMP / 8, blk, 0, stream>>>(outa, evi, alpha);
}